// MovingNormalizationLayer_5927054868901
// MI455X (gfx1250) — compile-verified
//
#include <hip/hip_runtime.h>
#include <cstdint>
#include <cstddef>

// Moving normalization along W (window 100, zero 'same' padding) for
// (256, 256, 2000) fp32. One 256-thread block per row.
//   - Row staged into LDS with CDNA5 async global->LDS copies (ASYNCcnt).
//   - Block-wide f32 prefix sums of x and x^2 (same algorithm as reference's
//     cumsum), wave32 __shfl_up scan + cross-wave offsets.
//   - Prefix arrays stored skewed (idx + idx>>6) -> conflict-free LDS.
//   - O(1) window: sum = P[min(i+50,N)] - P[max(i-50,0)]; NT 128-bit stores.

#define N_W    2000
#define WIN    100
#define PAOFF  51          // P[L] lives at logical index L+51; [0,51] = zero halo
#define PA_SZ  2120        // max logical idx 2051 -> phys 2083; padded
#define FLT_MAX_V 3.402823466385288598117e38f

typedef float v4f __attribute__((ext_vector_type(4)));

__device__ __forceinline__ int sk(int i) { return i + (i >> 6); }  // bank skew

__global__ __launch_bounds__(256)
void movnorm3d_w100_kernel(const float* __restrict__ x, float* __restrict__ out)
{
    __shared__ __align__(16) float xs[N_W];      // raw row (async dest)
    __shared__ __align__(16) float pa1[PA_SZ];   // skewed prefix of x
    __shared__ __align__(16) float pa2[PA_SZ];   // skewed prefix of x*x
    __shared__ float wtot1[8], wtot2[8];         // per-wave scan totals

    const int t    = threadIdx.x;                // 0..255
    const int lane = t & 31;
    const int wv   = t >> 5;                     // 0..7
    const int row  = blockIdx.x;
    const float* xr   = x   + (size_t)row * N_W;
    float*       outr = out + (size_t)row * N_W;

    // Zero halo of prefix arrays: logical idx [0,51] (phys == logical here).
    if (t < 52) { pa1[t] = 0.0f; pa2[t] = 0.0f; }

    // ---- CDNA5 async copy: 8000 B row -> LDS, 500 x 16B chunks (streaming) ----
    {
        uint32_t laddr = (uint32_t)(uintptr_t)(&xs[0]) + 16u * (uint32_t)t;
        uint64_t gaddr = (uint64_t)(uintptr_t)xr + 16u * (uint32_t)t;
        asm volatile("global_load_async_to_lds_b128 %0, %1, off th:TH_LOAD_NT"
                     :: "v"(laddr), "v"(gaddr) : "memory");
        if (t < 244) {
            asm volatile("global_load_async_to_lds_b128 %0, %1, off offset:4096 th:TH_LOAD_NT"
                         :: "v"(laddr), "v"(gaddr) : "memory");
        }
        asm volatile("s_wait_asynccnt 0x0" ::: "memory");
    }
    __syncthreads();                             // xs ready

    // ---- Phase A: per-thread local inclusive scans (registers) ----
    const int i0 = t * 8;
    float xv[8];
    float l1[8], l2[8];
    float tot1 = 0.0f, tot2 = 0.0f;
    if (t < 250) {
        v4f a = *(const v4f*)(xs + i0);
        v4f b = *(const v4f*)(xs + i0 + 4);
        xv[0]=a.x; xv[1]=a.y; xv[2]=a.z; xv[3]=a.w;
        xv[4]=b.x; xv[5]=b.y; xv[6]=b.z; xv[7]=b.w;
#pragma unroll
        for (int q = 0; q < 8; ++q) {
            tot1 += xv[q];
            tot2 += xv[q] * xv[q];
            l1[q] = tot1;
            l2[q] = tot2;
        }
    } else {
#pragma unroll
        for (int q = 0; q < 8; ++q) { xv[q] = 0.0f; l1[q] = 0.0f; l2[q] = 0.0f; }
    }

    // ---- Phase B: wave32 inclusive scan of totals (registers only) ----
    float v1 = tot1, v2 = tot2;
#pragma unroll
    for (int d = 1; d < 32; d <<= 1) {
        float n1 = __shfl_up(v1, (unsigned)d, 32);
        float n2 = __shfl_up(v2, (unsigned)d, 32);
        if (lane >= d) { v1 += n1; v2 += n2; }
    }
    if (lane == 31) { wtot1[wv] = v1; wtot2[wv] = v2; }
    __syncthreads();                             // wave totals ready

    float wo1 = 0.0f, wo2 = 0.0f;
    for (int j = 0; j < wv; ++j) { wo1 += wtot1[j]; wo2 += wtot2[j]; }
    const float e1 = wo1 + v1 - tot1;            // exclusive prefix of thread t
    const float e2 = wo2 + v2 - tot2;

    // ---- Phase C: write P[L], L = i0+1 .. i0+8, at skewed idx (L+51) ----
    if (t < 250) {
#pragma unroll
        for (int q = 0; q < 8; ++q) {
            int idx = i0 + q + 1 + PAOFF;
            pa1[sk(idx)] = e1 + l1[q];
            pa2[sk(idx)] = e2 + l2[q];
        }
    }
    __syncthreads();                             // prefix arrays ready

    // ---- Phase D: O(1) windows, normalize, NT store ----
    if (t < 250) {
        const float inv_n = 1.0f / (float)WIN;
        float r[8];
#pragma unroll
        for (int q = 0; q < 8; ++q) {
            const int i = i0 + q;
            int lhi = i + 50;
            lhi = lhi > N_W ? N_W : lhi;         // P[min(i+50, N)]
            const float s1hi = pa1[sk(lhi + PAOFF)];
            const float s2hi = pa2[sk(lhi + PAOFF)];
            const float s1lo = pa1[sk(i + 1)];   // (i-50)+51; halo gives P[0]=0
            const float s2lo = pa2[sk(i + 1)];
            float mean  = (s1hi - s1lo) * inv_n;
            float mean2 = (s2hi - s2lo) * inv_n;
            float var   = mean2 - mean * mean;
            var = var > 0.0f ? var : 0.0f;       // clip(., 0)
            float sd = __builtin_sqrtf(var);
            float v  = (xv[q] - mean) / sd;
            // nan_to_num(nan=0): NaN->0, +inf->FLT_MAX, -inf->-FLT_MAX
            v = __builtin_isnan(v) ? 0.0f : v;
            v = v >  FLT_MAX_V ?  FLT_MAX_V : v;
            v = v < -FLT_MAX_V ? -FLT_MAX_V : v;
            r[q] = v;
        }
        v4f lo4 = { r[0], r[1], r[2], r[3] };
        v4f hi4 = { r[4], r[5], r[6], r[7] };
        __builtin_nontemporal_store(lo4, (v4f*)(outr + i0));
        __builtin_nontemporal_store(hi4, (v4f*)(outr + i0 + 4));
    }
}

extern "C" void kernel_launch(void* const* d_in, const int* in_sizes, int n_in,
                              void* d_out, int out_size, void* d_ws, size_t ws_size,
                              hipStream_t stream) {
    (void)n_in; (void)out_size; (void)d_ws; (void)ws_size;
    const float* x   = (const float*)d_in[0];
    float*       out = (float*)d_out;
    const int rows = in_sizes[0] / N_W;          // 256*256 = 65536
    movnorm3d_w100_kernel<<<dim3(rows), dim3(256), 0, stream>>>(x, out);
}